// graph_function_41910290874470
// MI455X (gfx1250) — compile-verified
//
#include <hip/hip_runtime.h>
#include <hip/hip_bf16.h>

#define HIDDEN 128
#define BN_EPS 1e-5f

typedef float v2f __attribute__((ext_vector_type(2)));
typedef float v4f __attribute__((ext_vector_type(4)));
typedef float v8f __attribute__((ext_vector_type(8)));

// ---------------- degree / dinv ----------------

__global__ void k_init_deg(float* deg, int n) {
    int i = blockIdx.x * blockDim.x + threadIdx.x;
    if (i < n) deg[i] = 1.0f;   // self-loop contributes 1 to every node's degree
}

__global__ void k_deg_edges(const int* __restrict__ ei, float* deg, int E) {
    int e = blockIdx.x * blockDim.x + threadIdx.x;
    if (e < E) {
        int c = ei[E + e];      // col = target node
        atomicAdd(&deg[c], 1.0f);
    }
}

__global__ void k_rsqrt_inplace(float* deg, int n) {
    int i = blockIdx.x * blockDim.x + threadIdx.x;
    if (i < n) deg[i] = rsqrtf(deg[i]);   // deg >= 1 always (self-loops)
}

// ---------------- h = relu(x) @ W  via V_WMMA_F32_16X16X4_F32 ----------------
// Layouts per CDNA5 ISA 7.12.2 (wave32):
//   A 16x4 f32 : lane l holds A[m=l&15][kb], A[m][kb+1], kb = 2*(l>>4)
//   B 4x16 f32 : lane l holds B[kb][n=l&15], B[kb+1][n]   (mirrored striping)
//   C/D 16x16  : VGPR r, lane l -> D[r + 8*(l>>4)][l&15]
// Block = 256 threads = 8 waves; block owns 16 node-rows; wave w owns cols [16w,16w+16).
__global__ void __launch_bounds__(256)
k_gemm_relu_wmma(const float* __restrict__ x, const float* __restrict__ W,
                 float* __restrict__ h) {
    const int lane = threadIdx.x & 31;
    const int wave = threadIdx.x >> 5;
    const int m16  = lane & 15;
    const int grp  = lane >> 4;          // 0 or 1
    const int node0 = blockIdx.x * 16;   // 100000 = 6250 * 16 exactly
    const int n0    = wave * 16;

    const int m = node0 + m16;           // A row this lane loads
    const int n = n0 + m16;              // B/D column this lane owns

    v8f c = {};

    #pragma unroll
    for (int it = 0; it < HIDDEN / 4; ++it) {
        const int kb = it * 4 + 2 * grp;
        // A fragment: two consecutive K values from row m, with ReLU
        v2f a = *(const v2f*)(x + (size_t)m * HIDDEN + kb);
        a.x = fmaxf(a.x, 0.0f);
        a.y = fmaxf(a.y, 0.0f);
        // B fragment: W[kb][n], W[kb+1][n]
        v2f b;
        b.x = W[(size_t)kb * HIDDEN + n];
        b.y = W[(size_t)(kb + 1) * HIDDEN + n];
        c = __builtin_amdgcn_wmma_f32_16x16x4_f32(
                /*neg_a=*/false, a, /*neg_b=*/false, b,
                /*c_mod=*/(short)0, c, /*reuse_a=*/false, /*reuse_b=*/false);
    }

    // Scatter D tile to h[node0..node0+15][n0..n0+15]
    #pragma unroll
    for (int r = 0; r < 8; ++r) {
        const int mr = node0 + r + 8 * grp;
        h[(size_t)mr * HIDDEN + n] = c[r];
    }
}

// ---------- out = dinv[i]^2 * h[i] + b  (self-loop + conv bias), float4 ----------

__global__ void k_init_out(const float* __restrict__ h, const float* __restrict__ dinv,
                           const float* __restrict__ bias, float* __restrict__ out, int n) {
    int idx = blockIdx.x * blockDim.x + threadIdx.x;   // over n * 32 (float4 groups)
    if (idx < n * (HIDDEN / 4)) {
        int i  = idx >> 5;                 // node
        int f4 = (idx & 31) << 2;          // first feature of this group
        float d  = dinv[i];
        float d2 = d * d;
        v4f hv = *(const v4f*)(h + ((size_t)i * HIDDEN + f4));
        v4f bv = *(const v4f*)(bias + f4);
        v4f o  = d2 * hv + bv;
        *(v4f*)(out + ((size_t)i * HIDDEN + f4)) = o;
    }
}

// ---- edge scatter: out[col] += dinv[row]*dinv[col]*h[row]; 32 lanes/edge, b128 gathers ----

__global__ void k_edge_scatter(const int* __restrict__ ei, const float* __restrict__ h,
                               const float* __restrict__ dinv, float* __restrict__ out, int E) {
    long long idx = (long long)blockIdx.x * blockDim.x + threadIdx.x;  // over E * 32
    if (idx < (long long)E * (HIDDEN / 4)) {
        int e  = (int)(idx >> 5);          // one wave handles one edge
        int f4 = (int)(idx & 31) << 2;     // 4 consecutive features per lane
        int r = ei[e];                     // wave-uniform -> broadcast load
        int c = ei[E + e];
        float s = dinv[r] * dinv[c];
        v4f hv = *(const v4f*)(h + ((size_t)r * HIDDEN + f4));  // coalesced b128
        float* o = out + ((size_t)c * HIDDEN + f4);
        atomicAdd(o + 0, s * hv.x);
        atomicAdd(o + 1, s * hv.y);
        atomicAdd(o + 2, s * hv.z);
        atomicAdd(o + 3, s * hv.w);
    }
}

// ---------------- BatchNorm statistics ----------------

__global__ void k_zero_sums(float* sums) {  // sums[0..127]=sum, sums[128..255]=sumsq
    int i = threadIdx.x;
    if (i < 2 * HIDDEN) sums[i] = 0.0f;
}

// block = 128 threads (one feature each), grid = nPart blocks; rows strided by nPart.
__global__ void __launch_bounds__(128)
k_bn_stats(const float* __restrict__ out, float* __restrict__ sums, int n, int nPart) {
    int f = threadIdx.x;
    float s = 0.0f, sq = 0.0f;
    for (int i = blockIdx.x; i < n; i += nPart) {
        float v = out[(size_t)i * HIDDEN + f];   // coalesced 512B row per block-iter
        s += v;
        sq += v * v;
    }
    atomicAdd(&sums[f], s);
    atomicAdd(&sums[HIDDEN + f], sq);
}

__global__ void k_bn_apply(float* __restrict__ out, const float* __restrict__ sums,
                           const float* __restrict__ gamma, const float* __restrict__ beta,
                           int n) {
    int idx = blockIdx.x * blockDim.x + threadIdx.x;   // over n * 32 (float4 groups)
    if (idx < n * (HIDDEN / 4)) {
        int i  = idx >> 5;
        int f4 = (idx & 31) << 2;
        float inv_n = 1.0f / (float)n;
        v4f sv  = *(const v4f*)(sums + f4);
        v4f sqv = *(const v4f*)(sums + HIDDEN + f4);
        v4f gv  = *(const v4f*)(gamma + f4);
        v4f bv  = *(const v4f*)(beta + f4);
        v4f v   = *(const v4f*)(out + ((size_t)i * HIDDEN + f4));
        v4f res;
        #pragma unroll
        for (int j = 0; j < 4; ++j) {
            float mean = sv[j] * inv_n;
            float var  = sqv[j] * inv_n - mean * mean;   // biased, like jnp.var
            res[j] = gv[j] * (v[j] - mean) * rsqrtf(var + BN_EPS) + bv[j];
        }
        *(v4f*)(out + ((size_t)i * HIDDEN + f4)) = res;
    }
}

// ---------------- launch ----------------

extern "C" void kernel_launch(void* const* d_in, const int* in_sizes, int n_in,
                              void* d_out, int out_size, void* d_ws, size_t ws_size,
                              hipStream_t stream) {
    const float* x     = (const float*)d_in[0];
    const int*   ei    = (const int*)d_in[1];
    const float* W     = (const float*)d_in[2];
    const float* bias  = (const float*)d_in[3];
    const float* gamma = (const float*)d_in[4];
    const float* beta  = (const float*)d_in[5];
    float* out = (float*)d_out;

    const int N = in_sizes[0] / HIDDEN;       // 100000
    const int E = in_sizes[1] / 2;            // 600000

    // workspace layout (floats): h[N*128] | dinv[N] | sums[256]
    float* h    = (float*)d_ws;
    float* dinv = h + (size_t)N * HIDDEN;
    float* sums = dinv + N;

    const int T = 256;

    // 1) degree with self-loops -> dinv
    k_init_deg<<<(N + T - 1) / T, T, 0, stream>>>(dinv, N);
    k_deg_edges<<<(E + T - 1) / T, T, 0, stream>>>(ei, dinv, E);
    k_rsqrt_inplace<<<(N + T - 1) / T, T, 0, stream>>>(dinv, N);

    // 2) h = relu(x) @ W   (WMMA fp32)
    k_gemm_relu_wmma<<<N / 16, 256, 0, stream>>>(x, W, h);

    // 3) out = dinv^2 * h + b, then edge scatter-add (float4 paths)
    long long nGroups = (long long)N * (HIDDEN / 4);
    k_init_out<<<(int)((nGroups + T - 1) / T), T, 0, stream>>>(h, dinv, bias, out, N);
    long long eGroups = (long long)E * (HIDDEN / 4);
    k_edge_scatter<<<(int)((eGroups + T - 1) / T), T, 0, stream>>>(ei, h, dinv, out, E);

    // 4) BatchNorm (training stats over nodes)
    const int NPART = 512;
    k_zero_sums<<<1, 256, 0, stream>>>(sums);
    k_bn_stats<<<NPART, 128, 0, stream>>>(out, sums, N, NPART);
    k_bn_apply<<<(int)((nGroups + T - 1) / T), T, 0, stream>>>(out, sums, gamma, beta, N);
}